// GridBasedNavierStokesLoss_14929306321126
// MI455X (gfx1250) — compile-verified
//
#include <hip/hip_runtime.h>

// ---------------------------------------------------------------------------
// GridBasedNavierStokesLoss on MI455X (gfx1250, wave32).
// Bandwidth-bound 7-point stencil + deterministic 2-stage reduction.
// Wave reduction uses V_WMMA_F32_16X16X4_F32 (A = partials, B = ones).
// ---------------------------------------------------------------------------

namespace nsk {

constexpr int   GN      = 192;                 // grid edge
constexpr int   GNT     = GN * GN * GN;        // 7,077,888 points
constexpr int   THREADS = 256;                 // 8 waves / block
constexpr int   BLOCKS  = 1728;                // 1728*256*16 == GNT exactly
constexpr int   ITERS   = 16;
constexpr float EPS     = 1e-8f;

typedef __attribute__((ext_vector_type(2))) float v2f;
typedef __attribute__((ext_vector_type(8))) float v8f;

struct F3 { float x, y, z; };

__device__ __forceinline__ F3 ld3(const float* __restrict__ b, int idx) {
    F3 r;
    r.x = b[3 * idx + 0];
    r.y = b[3 * idx + 1];
    r.z = b[3 * idx + 2];
    return r;
}

// Exact half-wave sum via WMMA: returns, for lanes 0-15, the sum of x over
// lanes {0..7, 16..23}; for lanes 16-31, the sum over lanes {8..15, 24..31}.
// (A 16x4 layout: lane L<16 holds (M=L,K=0/1), lane L>=16 holds (M=L-16,K=2/3);
//  K=1/3 slots are zero.  B = ones => D[m][n] = x_m + x_{m+16}.)
__device__ __forceinline__ float wave_halfsum_wmma(float x) {
    v2f a; a[0] = x;    a[1] = 0.0f;
    v2f b; b[0] = 1.0f; b[1] = 1.0f;
    v8f c = {};
    c = __builtin_amdgcn_wmma_f32_16x16x4_f32(
            /*neg_a=*/false, a, /*neg_b=*/false, b,
            /*c_mod=*/(short)0, c, /*reuse_a=*/false, /*reuse_b=*/false);
    return ((c[0] + c[1]) + (c[2] + c[3])) + ((c[4] + c[5]) + (c[6] + c[7]));
}

__global__ __launch_bounds__(THREADS)
void ns_residual_kernel(const float* __restrict__ u,
                        const float* __restrict__ uprev,
                        const float* __restrict__ p,
                        const float* __restrict__ rho,
                        const float* __restrict__ nu,
                        const float* __restrict__ h,
                        const float* __restrict__ dt,
                        const float* __restrict__ grav,
                        float* __restrict__ partials) {
    // Uniform scalars (L1/L2-resident after first block touches them).
    const float hx = h[0], hy = h[1], hz = h[2];
    const float i2hx = 0.5f / hx, i2hy = 0.5f / hy, i2hz = 0.5f / hz;
    const float ihx2 = 1.0f / (hx * hx), ihy2 = 1.0f / (hy * hy), ihz2 = 1.0f / (hz * hz);
    const float idt  = 1.0f / dt[0];
    const float nu0  = nu[0];
    const float gx = grav[0], gy = grav[1], gz = grav[2];

    float mom = 0.0f, cont = 0.0f;

    const int stride = BLOCKS * THREADS;
    int f = blockIdx.x * THREADS + threadIdx.x;

    for (int it = 0; it < ITERS; ++it, f += stride) {
        // Stream-ahead prefetch of next iteration's center data (gfx1250
        // global_prefetch_b8; no counters, purely a cache hint).
        int fn = f + stride;
        if (fn < GNT) {
            __builtin_prefetch(u + 3 * fn, 0, 1);
            __builtin_prefetch(uprev + 3 * fn, 0, 1);
            __builtin_prefetch(p + fn, 0, 1);
            __builtin_prefetch(rho + fn, 0, 1);
        }

        const int k = f % GN;
        const int t = f / GN;
        const int j = t % GN;
        const int i = t / GN;

        const int ip = (i + 1 == GN) ? 0 : i + 1, im = (i == 0) ? GN - 1 : i - 1;
        const int jp = (j + 1 == GN) ? 0 : j + 1, jm = (j == 0) ? GN - 1 : j - 1;
        const int kp = (k + 1 == GN) ? 0 : k + 1, km = (k == 0) ? GN - 1 : k - 1;

        const int c0  = (i * GN + j) * GN + k;
        const int cxp = (ip * GN + j) * GN + k, cxm = (im * GN + j) * GN + k;
        const int cyp = (i * GN + jp) * GN + k, cym = (i * GN + jm) * GN + k;
        const int czp = (i * GN + j) * GN + kp, czm = (i * GN + j) * GN + km;

        const F3 uc  = ld3(u, c0);
        const F3 Uxp = ld3(u, cxp), Uxm = ld3(u, cxm);
        const F3 Uyp = ld3(u, cyp), Uym = ld3(u, cym);
        const F3 Uzp = ld3(u, czp), Uzm = ld3(u, czm);
        const F3 upv = ld3(uprev, c0);

        const float pxp = p[cxp], pxm = p[cxm];
        const float pyp = p[cyp], pym = p[cym];
        const float pzp = p[czp], pzm = p[czm];
        const float irho = 1.0f / (rho[c0] + EPS);

        // Central differences (periodic, matches jnp.roll convention).
        const float dux_dx = (Uxp.x - Uxm.x) * i2hx;
        const float duy_dx = (Uxp.y - Uxm.y) * i2hx;
        const float duz_dx = (Uxp.z - Uxm.z) * i2hx;
        const float dux_dy = (Uyp.x - Uym.x) * i2hy;
        const float duy_dy = (Uyp.y - Uym.y) * i2hy;
        const float duz_dy = (Uyp.z - Uym.z) * i2hy;
        const float dux_dz = (Uzp.x - Uzm.x) * i2hz;
        const float duy_dz = (Uzp.y - Uzm.y) * i2hz;
        const float duz_dz = (Uzp.z - Uzm.z) * i2hz;

        const float dp_dx = (pxp - pxm) * i2hx;
        const float dp_dy = (pyp - pym) * i2hy;
        const float dp_dz = (pzp - pzm) * i2hz;

        // Laplacians.
        const float lap_x = (Uxp.x - 2.0f * uc.x + Uxm.x) * ihx2
                          + (Uyp.x - 2.0f * uc.x + Uym.x) * ihy2
                          + (Uzp.x - 2.0f * uc.x + Uzm.x) * ihz2;
        const float lap_y = (Uxp.y - 2.0f * uc.y + Uxm.y) * ihx2
                          + (Uyp.y - 2.0f * uc.y + Uym.y) * ihy2
                          + (Uzp.y - 2.0f * uc.y + Uzm.y) * ihz2;
        const float lap_z = (Uxp.z - 2.0f * uc.z + Uxm.z) * ihx2
                          + (Uyp.z - 2.0f * uc.z + Uym.z) * ihy2
                          + (Uzp.z - 2.0f * uc.z + Uzm.z) * ihz2;

        // Convection.
        const float conv_x = uc.x * dux_dx + uc.y * dux_dy + uc.z * dux_dz;
        const float conv_y = uc.x * duy_dx + uc.y * duy_dy + uc.z * duy_dz;
        const float conv_z = uc.x * duz_dx + uc.y * duz_dy + uc.z * duz_dz;

        // Momentum residuals.
        const float Rx = (uc.x - upv.x) * idt + conv_x + dp_dx * irho - nu0 * lap_x - gx;
        const float Ry = (uc.y - upv.y) * idt + conv_y + dp_dy * irho - nu0 * lap_y - gy;
        const float Rz = (uc.z - upv.z) * idt + conv_z + dp_dz * irho - nu0 * lap_z - gz;

        const float Rc = dux_dx + duy_dy + duz_dz;

        mom  += Rx * Rx + Ry * Ry + Rz * Rz;
        cont += Rc * Rc;
    }

    // ---- wave reduction via WMMA (EXEC is all-ones here: no divergence) ----
    const float sm = wave_halfsum_wmma(mom);
    const float sc = wave_halfsum_wmma(cont);

    __shared__ float lds_m[16];
    __shared__ float lds_c[16];
    const int lane = threadIdx.x & 31;
    const int wid  = threadIdx.x >> 5;
    if (lane == 0)  { lds_m[wid * 2 + 0] = sm; lds_c[wid * 2 + 0] = sc; }
    if (lane == 16) { lds_m[wid * 2 + 1] = sm; lds_c[wid * 2 + 1] = sc; }
    __syncthreads();
    if (threadIdx.x == 0) {
        float tm = 0.0f, tc = 0.0f;
        #pragma unroll
        for (int q = 0; q < 16; ++q) { tm += lds_m[q]; tc += lds_c[q]; }
        partials[2 * blockIdx.x + 0] = tm;
        partials[2 * blockIdx.x + 1] = tc;
    }
}

__global__ __launch_bounds__(256)
void ns_reduce_kernel(const float* __restrict__ partials, float* __restrict__ out) {
    __shared__ float sm[256];
    __shared__ float sc[256];
    float tm = 0.0f, tc = 0.0f;
    for (int q = threadIdx.x; q < BLOCKS; q += 256) {
        tm += partials[2 * q + 0];
        tc += partials[2 * q + 1];
    }
    sm[threadIdx.x] = tm;
    sc[threadIdx.x] = tc;
    __syncthreads();
    for (int s = 128; s > 0; s >>= 1) {
        if (threadIdx.x < s) {
            sm[threadIdx.x] += sm[threadIdx.x + s];
            sc[threadIdx.x] += sc[threadIdx.x + s];
        }
        __syncthreads();
    }
    if (threadIdx.x == 0) {
        const float inv = 1.0f / (float)GNT;
        const float ml = sm[0] * inv;
        const float cl = sc[0] * inv;
        out[0] = 1.0f * ml + 10.0f * cl;   // total
        out[1] = ml;                        // momentum_loss
        out[2] = cl;                        // continuity_loss
    }
}

} // namespace nsk

extern "C" void kernel_launch(void* const* d_in, const int* in_sizes, int n_in,
                              void* d_out, int out_size, void* d_ws, size_t ws_size,
                              hipStream_t stream) {
    (void)in_sizes; (void)n_in; (void)out_size; (void)ws_size;
    const float* u     = (const float*)d_in[0];
    const float* uprev = (const float*)d_in[1];
    const float* p     = (const float*)d_in[2];
    const float* rho   = (const float*)d_in[3];
    const float* nu    = (const float*)d_in[4];
    const float* h     = (const float*)d_in[5];
    const float* dt    = (const float*)d_in[6];
    const float* grav  = (const float*)d_in[7];

    float* partials = (float*)d_ws;   // 1728 * 2 floats = 13.5 KB scratch

    nsk::ns_residual_kernel<<<nsk::BLOCKS, nsk::THREADS, 0, stream>>>(
        u, uprev, p, rho, nu, h, dt, grav, partials);
    nsk::ns_reduce_kernel<<<1, 256, 0, stream>>>(partials, (float*)d_out);
}